// StrideNet_32959579029630
// MI455X (gfx1250) — compile-verified
//
#include <hip/hip_runtime.h>
#include <hip/hip_bf16.h>

// Y = 2*X elementwise on 8192x8192 f32, with Y[0,1] negated.
// Memory-bound: 256 MB in + 256 MB out = 512 MB @ 23.3 TB/s -> ~22 us floor.
// Path: straight-line b128 non-temporal loads/stores (TH=NT, stream exceeds
// the 192 MB L2 and is touched once), 8 independent 16B vectors per thread
// for MLP, fully coalesced wave32 access, zero bounds guards on the hot path.

typedef float v4f __attribute__((ext_vector_type(4)));

// Exact-fit kernel: launcher guarantees nvec == gridDim.x * blockDim.x * 8.
__global__ __launch_bounds__(256) void scale2_neg01_exact(
    const v4f* __restrict__ in, v4f* __restrict__ out) {
  const unsigned bd   = 256u;
  const unsigned base = blockIdx.x * (bd * 8u) + threadIdx.x;

  v4f v0 = __builtin_nontemporal_load(in + base + 0u * bd);
  v4f v1 = __builtin_nontemporal_load(in + base + 1u * bd);
  v4f v2 = __builtin_nontemporal_load(in + base + 2u * bd);
  v4f v3 = __builtin_nontemporal_load(in + base + 3u * bd);
  v4f v4 = __builtin_nontemporal_load(in + base + 4u * bd);
  v4f v5 = __builtin_nontemporal_load(in + base + 5u * bd);
  v4f v6 = __builtin_nontemporal_load(in + base + 6u * bd);
  v4f v7 = __builtin_nontemporal_load(in + base + 7u * bd);

  v0 *= 2.0f; v1 *= 2.0f; v2 *= 2.0f; v3 *= 2.0f;
  v4 *= 2.0f; v5 *= 2.0f; v6 *= 2.0f; v7 *= 2.0f;

  // Flat element 1 == vector 0 component .y (row 0, col 1): negate.
  // Only block 0 / thread 0 hits this: one v_cmp+cndmask per wave.
  if (base == 0u) v0.y = -v0.y;

  __builtin_nontemporal_store(v0, out + base + 0u * bd);
  __builtin_nontemporal_store(v1, out + base + 1u * bd);
  __builtin_nontemporal_store(v2, out + base + 2u * bd);
  __builtin_nontemporal_store(v3, out + base + 3u * bd);
  __builtin_nontemporal_store(v4, out + base + 4u * bd);
  __builtin_nontemporal_store(v5, out + base + 5u * bd);
  __builtin_nontemporal_store(v6, out + base + 6u * bd);
  __builtin_nontemporal_store(v7, out + base + 7u * bd);
}

// Generic guarded fallback for sizes that don't tile exactly (not used for
// the 8192x8192 shape, but keeps the launcher correct for any size).
__global__ __launch_bounds__(256) void scale2_neg01_guarded(
    const v4f* __restrict__ in, v4f* __restrict__ out, unsigned nvec) {
  const unsigned bd   = blockDim.x;
  const unsigned base = blockIdx.x * (bd * 4u) + threadIdx.x;
  #pragma unroll
  for (int k = 0; k < 4; ++k) {
    unsigned i = base + (unsigned)k * bd;
    if (i < nvec) {
      v4f v = __builtin_nontemporal_load(in + i);
      v *= 2.0f;
      if (i == 0u) v.y = -v.y;
      __builtin_nontemporal_store(v, out + i);
    }
  }
}

// Scalar tail for element counts not divisible by 4 (never hit here).
__global__ void scale2_tail_kernel(const float* __restrict__ in,
                                   float* __restrict__ out,
                                   unsigned start, unsigned n) {
  unsigned i = start + threadIdx.x;
  if (i < n) {
    float v = 2.0f * in[i];
    if (i == 1u) v = -v;
    out[i] = v;
  }
}

extern "C" void kernel_launch(void* const* d_in, const int* in_sizes, int n_in,
                              void* d_out, int out_size, void* d_ws, size_t ws_size,
                              hipStream_t stream) {
  (void)n_in; (void)d_ws; (void)ws_size; (void)in_sizes;

  const float* x = (const float*)d_in[0];
  float*       y = (float*)d_out;

  const unsigned n    = (unsigned)out_size;     // 67,108,864 floats
  const unsigned nvec = n >> 2;                 // 16,777,216 float4 vectors
  const unsigned tail = n & 3u;

  constexpr unsigned kBlock = 256;              // 8 waves (wave32) per block

  constexpr unsigned kVecPerBlkExact = kBlock * 8;  // 2048 vectors = 32 KB/block
  if (nvec != 0 && (nvec % kVecPerBlkExact) == 0u) {
    // 8192x8192 path: 8192 blocks, zero guards, straight-line clause-able
    // NT b128 load/store stream.
    const unsigned grid = nvec / kVecPerBlkExact;   // 8192
    scale2_neg01_exact<<<grid, kBlock, 0, stream>>>((const v4f*)x, (v4f*)y);
  } else if (nvec != 0) {
    constexpr unsigned kVecPerBlk = kBlock * 4;
    const unsigned grid = (nvec + kVecPerBlk - 1) / kVecPerBlk;
    scale2_neg01_guarded<<<grid, kBlock, 0, stream>>>((const v4f*)x, (v4f*)y, nvec);
  }
  if (tail > 0) {
    scale2_tail_kernel<<<1, 4, 0, stream>>>(x, y, nvec << 2, n);
  }
}